// ANNSimple_88570815578864
// MI455X (gfx1250) — compile-verified
//
#include <hip/hip_runtime.h>

typedef float v2f __attribute__((ext_vector_type(2)));
typedef float v8f __attribute__((ext_vector_type(8)));

#define KNOTS 66
#define TILE_ROWS 128

// ---------------------------------------------------------------------------
// Stage 1: build the two constant 66-vectors R_i, R_j (padded to 68) such that
// r_t = R_t . y  for every row y.  Single thread; double precision internally.
// ---------------------------------------------------------------------------
__global__ void spline_weights_kernel(const float* __restrict__ tI,
                                      const float* __restrict__ tJ,
                                      float* __restrict__ Rout) {
    if (threadIdx.x != 0 || blockIdx.x != 0) return;
    const int K = KNOTS;
    float xkf[KNOTS];
    for (int p = 0; p < K; ++p) xkf[p] = (float)((double)p / 65.0);
    xkf[K - 1] = 1.0f;
    double dx[KNOTS - 1];
    for (int p = 0; p < K - 1; ++p) dx[p] = (double)(xkf[p + 1] - xkf[p]);

    // Tridiagonal A (not-a-knot), row arrays
    double lo[KNOTS], di[KNOTS], up[KNOTS];
    lo[0] = 0.0; di[0] = dx[1]; up[0] = (double)(xkf[2] - xkf[0]);
    for (int p = 1; p < K - 1; ++p) {
        lo[p] = dx[p];
        di[p] = 2.0 * (dx[p] + dx[p - 1]);
        up[p] = dx[p - 1];
    }
    lo[K - 1] = (double)(xkf[K - 1] - xkf[K - 3]);
    di[K - 1] = dx[K - 3];
    up[K - 1] = 0.0;

    // Solve A^T w = e_c  (Thomas).  A^T: lower'[p]=up[p-1], upper'[p]=lo[p+1]
    auto solveT = [&](int c, double* w) {
        double cp[KNOTS], dp[KNOTS];
        double beta = di[0];
        cp[0] = lo[1] / beta;
        dp[0] = ((c == 0) ? 1.0 : 0.0) / beta;
        for (int p = 1; p < K; ++p) {
            double a = up[p - 1];
            double cc = (p < K - 1) ? lo[p + 1] : 0.0;
            double mlt = di[p] - a * cp[p - 1];
            cp[p] = cc / mlt;
            double rhs = ((p == c) ? 1.0 : 0.0) - a * dp[p - 1];
            dp[p] = rhs / mlt;
        }
        w[K - 1] = dp[K - 1];
        for (int p = K - 2; p >= 0; --p) w[p] = dp[p] - cp[p] * w[p + 1];
    };

    // G_c[p]: coefficient of y[p] in s_c
    auto computeG = [&](int c, double* G) {
        double w[KNOTS];
        solveT(c, w);
        double g[KNOTS - 1];
        double d02 = (double)(xkf[2] - xkf[0]);
        double dl  = (double)(xkf[K - 1] - xkf[K - 3]);
        double dx0 = dx[0], dx1 = dx[1], dxm1 = dx[K - 2], dxm2 = dx[K - 3];
        for (int m = 0; m < K - 1; ++m) {
            double gv = 0.0;
            if (m <= K - 3) gv += 3.0 * dx[m + 1] * w[m + 1];
            if (m >= 1)     gv += 3.0 * dx[m - 1] * w[m];
            if (m == 0)     gv += w[0] * ((dx0 + 2.0 * d02) * dx1 / d02);
            if (m == 1)     gv += w[0] * (dx0 * dx0 / d02);
            if (m == K - 3) gv += w[K - 1] * (dxm1 * dxm1 / dl);
            if (m == K - 2) gv += w[K - 1] * ((2.0 * dl + dxm1) * dxm2 / dl);
            g[m] = gv;
        }
        for (int p = 0; p < K; ++p) {
            double v = 0.0;
            if (p >= 1)     v += g[p - 1] / dx[p - 1];
            if (p <= K - 2) v -= g[p] / dx[p];
            G[p] = v;
        }
    };

    auto buildR = [&](float t, float* R) {
        int cnt = 0;
        for (int p = 0; p < K; ++p) if (xkf[p] <= t) cnt++;
        int k = cnt - 1;
        if (k < 0) k = 0;
        if (k > K - 2) k = K - 2;
        double G0[KNOTS], G1[KNOTS];
        computeG(k, G0);
        computeG(k + 1, G1);
        double h = (double)(xkf[k + 1] - xkf[k]);
        double u = (double)(t - xkf[k]);
        double u2 = u * u, u3 = u2 * u, h2 = h * h;
        double As = u3 / h2 - 2.0 * u2 / h + u;   // coeff on s_k
        double Bs = u3 / h2 - u2 / h;             // coeff on s_{k+1}
        double Cm = 3.0 * u2 / h - 2.0 * u3 / h2; // coeff on slope_k
        for (int p = 0; p < 68; ++p) {
            double v = 0.0;
            if (p < K) v = As * G0[p] + Bs * G1[p];
            if (p == k)     v += 1.0 - Cm / h;
            if (p == k + 1) v += Cm / h;
            R[p] = (float)v;
        }
    };

    buildR(tI[0], Rout);
    buildR(tJ[0], Rout + 68);
}

// ---------------------------------------------------------------------------
// Stage 2: streaming WMMA kernel.  Each 256-thread block stages 128 rows of x
// into LDS (unpadded stride 66 -> pure b128 memcpy from global).  Each wave
// runs 17 chained V_WMMA_F32_16X16X4_F32 (A = 16 rows x 68 cols of x,
// B panel = 16x68 in LDS, rows 0/1 = R_i/R_j, rows 2..15 zero -> branchless).
// K-columns 66/67 carry neighbor-row garbage but R[66]=R[67]=0.
// ---------------------------------------------------------------------------
__global__ __launch_bounds__(256) void spline_eval_wmma(
        const float* __restrict__ x, const float* __restrict__ Rw,
        float* __restrict__ ndis_out, float* __restrict__ ndis_ws, int nrows) {
    __shared__ __align__(16) float tile[TILE_ROWS * KNOTS + 8];
    __shared__ __align__(16) float Rl[16 * 68];
    __shared__ float ri[TILE_ROWS];
    __shared__ float rj[TILE_ROWS];

    const int tid = threadIdx.x;
    const int row0 = blockIdx.x * TILE_ROWS;

    // B panel: rows 0/1 from workspace, rows 2..15 zero (branchless B loads)
    for (int idx = tid; idx < 16 * 68; idx += 256)
        Rl[idx] = (idx < 2 * 68) ? Rw[idx] : 0.0f;

    // Contiguous b128 copy of the 128x66 tile (8448 floats = 2112 float4)
    const float4* src4 = (const float4*)(x + (size_t)row0 * KNOTS);
    float4* dst4 = (float4*)tile;
    for (int idx = tid; idx < (TILE_ROWS * KNOTS) / 4; idx += 256)
        dst4[idx] = src4[idx];
    __syncthreads();

    const int wave = tid >> 5;
    const int lane = tid & 31;
    const int m    = lane & 15;   // A: row / B,D: column index
    const int half = lane >> 4;   // K-half selector (A/B), M-half (D)
    const int arow = (wave * 16 + m) * KNOTS;
    const int brow = m * 68;

    v8f acc = {0.f, 0.f, 0.f, 0.f, 0.f, 0.f, 0.f, 0.f};
    #pragma unroll
    for (int kc = 0; kc < 17; ++kc) {
        const int kk = 4 * kc + 2 * half;
        v2f a, b;
        a.x = tile[arow + kk];
        a.y = tile[arow + kk + 1];
        b.x = Rl[brow + kk];
        b.y = Rl[brow + kk + 1];
        acc = __builtin_amdgcn_wmma_f32_16x16x4_f32(
            /*neg_a=*/false, a, /*neg_b=*/false, b,
            /*c_mod=*/(short)0, acc, /*reuse_a=*/false, /*reuse_b=*/false);
    }

    // D layout: VGPR g, lanes 0-15 -> M=g, lanes 16-31 -> M=g+8; column N=m.
    if (m == 0) {
        #pragma unroll
        for (int g = 0; g < 8; ++g) ri[wave * 16 + 8 * half + g] = acc[g];
    }
    if (m == 1) {
        #pragma unroll
        for (int g = 0; g < 8; ++g) rj[wave * 16 + 8 * half + g] = acc[g];
    }
    __syncthreads();

    if (tid < TILE_ROWS) {
        int row = row0 + tid;
        if (row < nrows) {
            float a_ = ri[tid], b_ = rj[tid];
            float nv = (a_ - b_) / (a_ + b_);
            ndis_out[row] = nv;   // output slot (d_out+1, unaligned ok for b32)
            ndis_ws[row]  = nv;   // aligned workspace copy for stage 3
        }
    }
}

// ---------------------------------------------------------------------------
// Stage 3: deterministic partial reduction of W1@ndis (10 dots) + sum(soc),
// sum(soc^2), all streams read as float4.  One 12-float partial per block.
// ---------------------------------------------------------------------------
__global__ __launch_bounds__(256) void mlp_partial(
        const float* __restrict__ ndis, const float* __restrict__ W1,
        const float* __restrict__ soc, float* __restrict__ partials, int N_) {
    float acc[10];
    #pragma unroll
    for (int r = 0; r < 10; ++r) acc[r] = 0.0f;
    float s1 = 0.0f, s2 = 0.0f;

    const int n4 = N_ >> 2;                  // N divisible by 4
    const float4* nd4  = (const float4*)ndis;
    const float4* soc4 = (const float4*)soc;
    const float4* W14  = (const float4*)W1;

    const int stride = gridDim.x * blockDim.x;
    for (int n = blockIdx.x * blockDim.x + threadIdx.x; n < n4; n += stride) {
        float4 v  = nd4[n];
        float4 sv = soc4[n];
        #pragma unroll
        for (int r = 0; r < 10; ++r) {
            float4 w = W14[(size_t)r * n4 + n];
            acc[r] += w.x * v.x + w.y * v.y + w.z * v.z + w.w * v.w;
        }
        s1 += sv.x + sv.y + sv.z + sv.w;
        s2 += sv.x * sv.x + sv.y * sv.y + sv.z * sv.z + sv.w * sv.w;
    }

    __shared__ float red[256];
    for (int j = 0; j < 12; ++j) {
        float v = (j < 10) ? acc[j] : ((j == 10) ? s1 : s2);
        red[threadIdx.x] = v;
        __syncthreads();
        for (int off = 128; off > 0; off >>= 1) {
            if (threadIdx.x < off) red[threadIdx.x] += red[threadIdx.x + off];
            __syncthreads();
        }
        if (threadIdx.x == 0) partials[blockIdx.x * 12 + j] = red[0];
        __syncthreads();
    }
}

// ---------------------------------------------------------------------------
// Stage 4: final reduction + MLP head + loss.  loss = N*sh^2 - 2*sh*Σsoc + Σsoc²
// ---------------------------------------------------------------------------
__global__ void mlp_final(const float* __restrict__ partials, int nparts,
                          const float* __restrict__ b1,
                          const float* __restrict__ W2,
                          const float* __restrict__ b2,
                          float* __restrict__ out_soc_hat,
                          float* __restrict__ out_loss, int N_) {
    if (threadIdx.x != 0 || blockIdx.x != 0) return;
    float sums[12];
    for (int j = 0; j < 12; ++j) {
        float a = 0.0f;
        for (int p = 0; p < nparts; ++p) a += partials[p * 12 + j];
        sums[j] = a;
    }
    float sh = 0.0f;
    for (int r = 0; r < 10; ++r) {
        float h = sums[r] + b1[r];
        h = (h > 0.0f) ? h : 0.01f * h;
        sh += W2[r] * h;
    }
    sh += b2[0];
    *out_soc_hat = sh;
    float loss = (float)N_ * sh * sh - 2.0f * sh * sums[10] + sums[11];
    *out_loss = loss;
}

// ---------------------------------------------------------------------------
extern "C" void kernel_launch(void* const* d_in, const int* in_sizes, int n_in,
                              void* d_out, int out_size, void* d_ws, size_t ws_size,
                              hipStream_t stream) {
    const float* x   = (const float*)d_in[0];
    const float* soc = (const float*)d_in[1];
    const float* ti  = (const float*)d_in[2];
    const float* tj  = (const float*)d_in[3];
    const float* W1  = (const float*)d_in[4];
    const float* b1  = (const float*)d_in[5];
    const float* W2  = (const float*)d_in[6];
    const float* b2  = (const float*)d_in[7];
    float* out = (float*)d_out;

    const int Nrows = in_sizes[1];           // soc has shape (N,)
    float* Rw       = (float*)d_ws;          // 2 x 68 floats
    float* partials = Rw + 256;              // 512 x 12 floats
    float* ndis_ws  = Rw + 8192;             // N floats, 16B-aligned

    spline_weights_kernel<<<1, 1, 0, stream>>>(ti, tj, Rw);

    int nblocks = (Nrows + TILE_ROWS - 1) / TILE_ROWS;
    spline_eval_wmma<<<nblocks, 256, 0, stream>>>(x, Rw, out + 1, ndis_ws, Nrows);

    mlp_partial<<<512, 256, 0, stream>>>(ndis_ws, W1, soc, partials, Nrows);

    mlp_final<<<1, 1, 0, stream>>>(partials, 512, b1, W2, b2,
                                   out, out + 1 + Nrows, Nrows);
}